// QuantizationLayerVoxGrid_27410481283598
// MI455X (gfx1250) — compile-verified
//
#include <hip/hip_runtime.h>
#include <hip/hip_bf16.h>

#define VC 9
#define VH 128
#define VW 128
#define MAXB 64   // LDS slots for batch maxima (reference uses B=32)

// ---------------------------------------------------------------------------
// Kernel 0: zero the output voxel grid (b128 stores) + tmax scratch in d_ws.
// ---------------------------------------------------------------------------
__global__ void vox_zero_kernel(float* __restrict__ out, int out_n,
                                unsigned* __restrict__ ws_max) {
    const int gid    = blockIdx.x * blockDim.x + threadIdx.x;
    const int stride = gridDim.x * blockDim.x;
    const int n4 = out_n >> 2;
    float4 z = make_float4(0.f, 0.f, 0.f, 0.f);
    float4* __restrict__ out4 = reinterpret_cast<float4*>(out);
    for (int j = gid; j < n4; j += stride) out4[j] = z;
    for (int j = (n4 << 2) + gid; j < out_n; j += stride) out[j] = 0.f;
    if (gid < MAXB) ws_max[gid] = 0u;
}

// Unpack a group of 4 events (5 x float4 = 80B, 16B-aligned for any group id).
__device__ __forceinline__ void load_group4(const float4* __restrict__ ev4,
                                            long long g, float f[20]) {
    const float4 a = ev4[5 * g + 0];
    const float4 b = ev4[5 * g + 1];
    const float4 c = ev4[5 * g + 2];
    const float4 d = ev4[5 * g + 3];
    const float4 e = ev4[5 * g + 4];
    f[0]=a.x; f[1]=a.y; f[2]=a.z; f[3]=a.w;
    f[4]=b.x; f[5]=b.y; f[6]=b.z; f[7]=b.w;
    f[8]=c.x; f[9]=c.y; f[10]=c.z; f[11]=c.w;
    f[12]=d.x; f[13]=d.y; f[14]=d.z; f[15]=d.w;
    f[16]=e.x; f[17]=e.y; f[18]=e.z; f[19]=e.w;
}

// ---------------------------------------------------------------------------
// Kernel 1: per-batch segment max of t.  Events are batch-sorted, so each
// thread walks a contiguous chunk of 4-event groups keeping a register max,
// flushing to LDS atomicMax only on batch transitions (positive floats
// compare correctly as uint bit patterns).
// ---------------------------------------------------------------------------
__global__ void vox_tmax_kernel(const float* __restrict__ ev, int n,
                                unsigned* __restrict__ ws_max,
                                const int* __restrict__ nb_ptr) {
    __shared__ unsigned smax[MAXB];
    const int tid = threadIdx.x;
    const int nb  = *nb_ptr;
    if (tid < MAXB) smax[tid] = 0u;
    __syncthreads();

    const float4* __restrict__ ev4 = reinterpret_cast<const float4*>(ev);
    const long long total   = (long long)gridDim.x * blockDim.x;
    const long long gid     = (long long)blockIdx.x * blockDim.x + tid;
    const long long ngroups = n >> 2;          // groups of 4 events
    const long long per     = (ngroups + total - 1) / total;
    long long g   = gid * per;
    long long end = g + per; if (end > ngroups) end = ngroups;

    int   cur = -1;
    float m   = 0.f;
    float f[20];
    for (; g < end; ++g) {
        load_group4(ev4, g, f);
        #pragma unroll
        for (int j = 0; j < 4; ++j) {
            const float t = f[5 * j + 2];
            const int   b = (int)f[5 * j + 4];
            if (b != cur) {
                if (cur >= 0 && cur < MAXB)
                    atomicMax(&smax[cur], __float_as_uint(m));
                cur = b;
                m   = t;
            } else {
                m = fmaxf(m, t);
            }
        }
    }
    if (cur >= 0 && cur < MAXB) atomicMax(&smax[cur], __float_as_uint(m));

    // tail events (n % 4), normally empty
    for (long long i = (ngroups << 2) + gid; i < n; i += total) {
        const float t = ev[5 * i + 2];
        const int   b = (int)ev[5 * i + 4];
        if (b >= 0 && b < MAXB) atomicMax(&smax[b], __float_as_uint(t));
    }

    __syncthreads();
    if (tid < nb && tid < MAXB) atomicMax(&ws_max[tid], smax[tid]);
}

// ---------------------------------------------------------------------------
// Kernel 2: scatter-add.  One global_atomic_add_f32 per valid event; events
// re-read from L2 (warmed by kernel 1).  4 events per iteration via b128
// loads; prefetch next grid-stride group (global_prefetch_b8).
// ---------------------------------------------------------------------------
__device__ __forceinline__ void scatter_one(const float f[5],
                                            const float* __restrict__ tmax_s,
                                            float* __restrict__ out,
                                            int num_voxels) {
    const float x = f[0], y = f[1], t = f[2], p = f[3];
    const int  bi = (int)f[4];
    const int  bl = (bi >= 0 && bi < MAXB) ? bi : 0;
    const float tn = t / tmax_s[bl];            // true division: match jnp exactly
    if (tn > 0.f && tn <= 1.f) {
        const int bin = (int)ceilf(tn * (float)VC) - 1;
        int idx = (int)x
                + VW * (int)y
                + VW * VH * bin
                + VW * VH * VC * (int)p
                + VW * VH * VC * 2 * bi;
        idx = min(max(idx, 0), num_voxels - 1);
        atomicAdd(&out[idx], 1.0f);              // global_atomic_add_f32 (no return)
    }
}

__global__ void vox_scatter_kernel(const float* __restrict__ ev, int n,
                                   const unsigned* __restrict__ ws_max,
                                   float* __restrict__ out, int num_voxels,
                                   const int* __restrict__ nb_ptr) {
    __shared__ float tmax_s[MAXB];
    const int nb = *nb_ptr;
    if (threadIdx.x < MAXB) {
        float tm = (threadIdx.x < nb) ? __uint_as_float(ws_max[threadIdx.x]) : 1.f;
        tmax_s[threadIdx.x] = (tm > 0.f) ? tm : 1.f;
    }
    __syncthreads();

    const float4* __restrict__ ev4 = reinterpret_cast<const float4*>(ev);
    const long long total   = (long long)gridDim.x * blockDim.x;
    const long long gid     = (long long)blockIdx.x * blockDim.x + threadIdx.x;
    const long long ngroups = n >> 2;

    float f[20];
    for (long long g = gid; g < ngroups; g += total) {
        const long long gnext = g + total;
        if (gnext < ngroups)
            __builtin_prefetch(&ev4[5 * gnext], 0, 0);   // global_prefetch_b8

        load_group4(ev4, g, f);
        #pragma unroll
        for (int j = 0; j < 4; ++j)
            scatter_one(&f[5 * j], tmax_s, out, num_voxels);
    }

    // tail events (n % 4), normally empty
    for (long long i = (ngroups << 2) + gid; i < n; i += total) {
        float ftail[5];
        #pragma unroll
        for (int k = 0; k < 5; ++k) ftail[k] = ev[5 * i + k];
        scatter_one(ftail, tmax_s, out, num_voxels);
    }
}

// ---------------------------------------------------------------------------
extern "C" void kernel_launch(void* const* d_in, const int* in_sizes, int n_in,
                              void* d_out, int out_size, void* d_ws, size_t ws_size,
                              hipStream_t stream) {
    const float* ev     = (const float*)d_in[0];
    const int*   nb_ptr = (const int*)d_in[1];
    float*       out    = (float*)d_out;
    unsigned*    wsm    = (unsigned*)d_ws;

    const int n = in_sizes[0] / 5;   // number of events

    const int threads = 256;         // 8 wave32s per block
    const int blocks  = 2048;

    vox_zero_kernel<<<blocks, threads, 0, stream>>>(out, out_size, wsm);
    vox_tmax_kernel<<<blocks, threads, 0, stream>>>(ev, n, wsm, nb_ptr);
    vox_scatter_kernel<<<blocks, threads, 0, stream>>>(ev, n, wsm, out, out_size, nb_ptr);
}